// DeepMOI_11630771437871
// MI455X (gfx1250) — compile-verified
//
#include <hip/hip_runtime.h>
#include <hip/hip_bf16.h>
#include <math.h>

// Problem constants (match reference)
#define BB   32
#define NN   5000
#define BN   160000      // BB*NN
#define INF_ 16
#define HH   64
#define EE   2560000
#define PP   300
#define LL   100

typedef __attribute__((ext_vector_type(2))) float v2f;
typedef __attribute__((ext_vector_type(8))) float v8f;

// ---------------------------------------------------------------------------
// zero fill (float4 granularity)
__global__ void zero_f4(float* __restrict__ p, int n4) {
    int i = blockIdx.x * blockDim.x + threadIdx.x;
    if (i < n4) ((float4*)p)[i] = make_float4(0.f, 0.f, 0.f, 0.f);
}

// ---------------------------------------------------------------------------
// segment_sum: agg[dst[e]] += X[src[e]], feature dim K (16 or 64).
// One thread per (edge, 4-float chunk); atomics resolve in L2 (has atomic units).
template<int K>
__global__ void scatter_add(const float* __restrict__ X,
                            const int* __restrict__ src,
                            const int* __restrict__ dst,
                            float* __restrict__ agg) {
    const int CH = K / 4;                  // chunks per row
    int tid = blockIdx.x * blockDim.x + threadIdx.x;
    if (tid >= EE * CH) return;
    int e = tid / CH;
    int c = (tid - e * CH) * 4;
    int s = src[e];
    int d = dst[e];
    float4 v = *(const float4*)(X + (size_t)s * K + c);
    float* out = agg + (size_t)d * K + c;
    atomicAdd(out + 0, v.x);
    atomicAdd(out + 1, v.y);
    atomicAdd(out + 2, v.z);
    atomicAdd(out + 3, v.w);
}

// ---------------------------------------------------------------------------
// Y = relu((X0 + AGG) @ W + bias), X0/AGG: [BN,K], W: [K,64], Y: [BN,64]
// f32 WMMA 16x16x4. One wave per 16-row tile; 4 N-tiles of 16; K/4 k-steps.
// A layout (16x4 f32): lane m=lane&15; VGPR0 = K(2*half), VGPR1 = K(2*half+1).
// B layout (4x16 f32): lane n=lane&15; VGPR0 = K(2*half) row, VGPR1 = next K.
// C/D (16x16 f32): VGPR v -> row m = v + 8*half, col n = lane&15.
template<int K>
__global__ void gin_gemm_relu(const float* __restrict__ X0,
                              const float* __restrict__ AGG,
                              const float* __restrict__ W,
                              const float* __restrict__ Bv,
                              float* __restrict__ Y) {
    const int lane = threadIdx.x & 31;
    const int wave = threadIdx.x >> 5;
    const int tile = blockIdx.x * (blockDim.x >> 5) + wave;  // 16-row tile id
    if (tile >= BN / 16) return;                             // wave-uniform

    const int half = lane >> 4;          // 0: lanes 0-15, 1: lanes 16-31
    const int mrow = (lane & 15);
    const int row  = tile * 16 + mrow;

    const float* x0 = X0  + (size_t)row * K;
    const float* ag = AGG + (size_t)row * K;

    // Load+fuse A tile (x = h + agg) for all k-steps
    v2f a[K / 4];
#pragma unroll
    for (int ks = 0; ks < K / 4; ++ks) {
        int k = ks * 4 + 2 * half;
        a[ks].x = x0[k]     + ag[k];
        a[ks].y = x0[k + 1] + ag[k + 1];
    }

#pragma unroll
    for (int nt = 0; nt < 4; ++nt) {
        const int n = nt * 16 + (lane & 15);
        const float bias = Bv[n];
        v8f c;
#pragma unroll
        for (int i = 0; i < 8; ++i) c[i] = bias;   // accumulate on top of bias

#pragma unroll
        for (int ks = 0; ks < K / 4; ++ks) {
            int k = ks * 4 + 2 * half;
            v2f b;
            b.x = W[(size_t)k * HH + n];
            b.y = W[(size_t)(k + 1) * HH + n];
            c = __builtin_amdgcn_wmma_f32_16x16x4_f32(
                    false, a[ks], false, b, (short)0, c, false, false);
        }
        // fused relu + store
#pragma unroll
        for (int v = 0; v < 8; ++v) {
            int m = tile * 16 + v + 8 * half;
            float val = c[v];
            Y[(size_t)m * HH + n] = val > 0.f ? val : 0.f;
        }
    }
}

// ---------------------------------------------------------------------------
// Per-graph mean over N nodes: mean[b][f] = (1/N) * sum_r h2[b*N+r][f]
__global__ void graph_mean(const float* __restrict__ h2, float* __restrict__ mean) {
    const int b = blockIdx.x;
    const int t = threadIdx.x;           // 256 threads
    const int f = t & 63;
    const int g = t >> 6;                // 4 row groups
    const float* base = h2 + (size_t)b * NN * HH;
    float acc = 0.f;
    for (int r = g; r < NN; r += 4) acc += base[(size_t)r * HH + f];
    __shared__ float red[256];
    red[t] = acc;
    __syncthreads();
    if (g == 0) {
        float v = red[f] + red[f + 64] + red[f + 128] + red[f + 192];
        mean[b * HH + f] = v * (1.0f / NN);
    }
}

// ---------------------------------------------------------------------------
// For each (b,p): sum-pool pathway nodes, concat with graph mean,
// dot with w_lin1, tanh -> s[b,p]. Block = (b*P+p), 64 threads (one per feature).
__global__ void pathway_score(const float* __restrict__ h2,
                              const int* __restrict__ pathway,
                              const float* __restrict__ mean,
                              const float* __restrict__ w_lin1,
                              const float* __restrict__ b_lin1,
                              float* __restrict__ s) {
    const int bp = blockIdx.x;
    const int b = bp / PP;
    const int p = bp - b * PP;
    const int f = threadIdx.x;           // 64 threads
    const float* base = h2 + (size_t)b * NN * HH;
    const int* path = pathway + p * LL;
    float sum = 0.f;
    for (int l = 0; l < LL; ++l) {
        int node = path[l];              // scalar broadcast
        sum += base[(size_t)node * HH + f];
    }
    float contrib = mean[b * HH + f] * w_lin1[f] + sum * w_lin1[HH + f];
    __shared__ float red[64];
    red[f] = contrib;
    __syncthreads();
    if (f < 32) {
        float v = red[f] + red[f + 32];
        for (int off = 16; off; off >>= 1) v += __shfl_down(v, off, 32);
        if (f == 0) s[bp] = tanhf(v + b_lin1[0]);
    }
}

// ---------------------------------------------------------------------------
// out[b] = softmax(s[b,:] @ W_out + b_out). One wave per graph.
__global__ void head_softmax(const float* __restrict__ s,
                             const float* __restrict__ W_out,
                             const float* __restrict__ b_out,
                             float* __restrict__ out) {
    const int b = blockIdx.x;
    const int lane = threadIdx.x;        // 32
    float o0 = 0.f, o1 = 0.f;
    for (int p = lane; p < PP; p += 32) {
        float sv = s[b * PP + p];
        o0 += sv * W_out[p * 2 + 0];
        o1 += sv * W_out[p * 2 + 1];
    }
    for (int off = 16; off; off >>= 1) {
        o0 += __shfl_down(o0, off, 32);
        o1 += __shfl_down(o1, off, 32);
    }
    if (lane == 0) {
        o0 += b_out[0];
        o1 += b_out[1];
        float m = fmaxf(o0, o1);
        float e0 = __expf(o0 - m), e1 = __expf(o1 - m);
        float inv = 1.f / (e0 + e1);
        out[b * 2 + 0] = e0 * inv;
        out[b * 2 + 1] = e1 * inv;
    }
}

// ---------------------------------------------------------------------------
extern "C" void kernel_launch(void* const* d_in, const int* in_sizes, int n_in,
                              void* d_out, int out_size, void* d_ws, size_t ws_size,
                              hipStream_t stream) {
    const float* h       = (const float*)d_in[0];   // [BN,16]
    const int*   src     = (const int*)  d_in[1];   // [E]
    const int*   dst     = (const int*)  d_in[2];   // [E]
    const int*   pathway = (const int*)  d_in[3];   // [P,L]
    // d_in[4] = num_graphs (compile-time constant BB)
    const float* W1      = (const float*)d_in[5];   // [16,64]
    const float* b1      = (const float*)d_in[6];   // [64]
    const float* W2      = (const float*)d_in[7];   // [64,64]
    const float* b2      = (const float*)d_in[8];   // [64]
    const float* w_lin1  = (const float*)d_in[9];   // [128,1]
    const float* b_lin1  = (const float*)d_in[10];  // [1]
    const float* W_out   = (const float*)d_in[11];  // [P,2]
    const float* b_out   = (const float*)d_in[12];  // [2]
    float* out = (float*)d_out;                     // [B,2]

    // Workspace layout (floats)
    float* ws   = (float*)d_ws;
    float* agg0 = ws;                         //  2,560,000
    float* agg1 = agg0 + (size_t)BN * INF_;   // 10,240,000
    float* h1   = agg1 + (size_t)BN * HH;     // 10,240,000
    float* h2   = h1   + (size_t)BN * HH;     // 10,240,000
    float* mean = h2   + (size_t)BN * HH;     //      2,048
    float* sbuf = mean + (size_t)BB * HH;     //      9,600

    // 1) zero the two atomic accumulators (agg0 + agg1 are contiguous)
    {
        int n4 = (BN * INF_ + BN * HH) / 4;   // 3,200,000 float4s
        zero_f4<<<(n4 + 255) / 256, 256, 0, stream>>>(agg0, n4);
    }
    // 2) agg0 = segment_sum(h[src], dst)
    {
        int nth = EE * (INF_ / 4);
        scatter_add<INF_><<<(nth + 255) / 256, 256, 0, stream>>>(h, src, dst, agg0);
    }
    // 3) h1 = relu((h + agg0) @ W1 + b1)   -- f32 WMMA
    gin_gemm_relu<INF_><<<BN / 16 / 8, 256, 0, stream>>>(h, agg0, W1, b1, h1);
    // 4) agg1 = segment_sum(h1[src], dst)
    {
        int nth = EE * (HH / 4);
        scatter_add<HH><<<(nth + 255) / 256, 256, 0, stream>>>(h1, src, dst, agg1);
    }
    // 5) h2 = relu((h1 + agg1) @ W2 + b2)  -- f32 WMMA
    gin_gemm_relu<HH><<<BN / 16 / 8, 256, 0, stream>>>(h1, agg1, W2, b2, h2);
    // 6) per-graph mean
    graph_mean<<<BB, 256, 0, stream>>>(h2, mean);
    // 7) pathway pooling + tanh head -> s[b,p]
    pathway_score<<<BB * PP, 64, 0, stream>>>(h2, pathway, mean, w_lin1, b_lin1, sbuf);
    // 8) s @ W_out + b_out, softmax -> out
    head_softmax<<<BB, 32, 0, stream>>>(sbuf, W_out, b_out, out);
}